// DGCNEncoder_2843268350769
// MI455X (gfx1250) — compile-verified
//
#include <hip/hip_runtime.h>

typedef __attribute__((ext_vector_type(2))) float v2f;
typedef __attribute__((ext_vector_type(8))) float v8f;
typedef __attribute__((ext_vector_type(4))) unsigned int u32x4;
typedef __attribute__((ext_vector_type(8))) int i32x8;
typedef __attribute__((ext_vector_type(4))) int i32x4;

#define NFEAT   128
#define PROJ    4
#define H2      132   // 128 + 4
#define LD      144   // padded row stride (multiple of 16)
#define OUTC    136   // 132 + 4
#define NT      9     // n-tiles = LD/16

static __device__ __forceinline__ void atomic_add_f32(float* p, float v) {
    __hip_atomic_fetch_add(p, v, __ATOMIC_RELAXED, __HIP_MEMORY_SCOPE_AGENT);
}

// ---------------------------------------------------------------- utilities
__global__ void zero_f4_kernel(float4* __restrict__ p, int n4) {
    int i = blockIdx.x * blockDim.x + threadIdx.x;
    if (i < n4) p[i] = make_float4(0.f, 0.f, 0.f, 0.f);
}

__global__ void degree_count_kernel(const int* __restrict__ col,
                                    float* __restrict__ deg, int E) {
    int e = blockIdx.x * blockDim.x + threadIdx.x;
    if (e < E) atomic_add_f32(&deg[col[e]], 1.0f);
}

// deg -> dinv (in place) and invdeg
__global__ void degree_fin_kernel(float* __restrict__ deg,
                                  float* __restrict__ invdeg, int n) {
    int i = blockIdx.x * blockDim.x + threadIdx.x;
    if (i < n) {
        float d = deg[i] + 1.0f;        // self loop
        deg[i]    = rsqrtf(d);
        invdeg[i] = 1.0f / d;
    }
}

// per-edge symmetric-norm coefficient
__global__ void coef_kernel(const int* __restrict__ row, const int* __restrict__ col,
                            const float* __restrict__ dinv, float* __restrict__ coef,
                            int E) {
    int e = blockIdx.x * blockDim.x + threadIdx.x;
    if (e < E) coef[e] = dinv[row[e]] * dinv[col[e]];
}

// K-pair interleaved, zero-padded weight staging:
// Wp[((r>>1)*ldp + c)*2 + (r&1)] = (c < cols) ? W[r*cols + c] : 0
// so rows {2j, 2j+1} of column c sit in one adjacent float pair (-> ds_load_b64).
__global__ void pad_w_kernel(const float* __restrict__ W, float* __restrict__ Wp,
                             int rows, int cols, int ldp) {
    int idx = blockIdx.x * blockDim.x + threadIdx.x;
    int total = rows * ldp;
    if (idx >= total) return;
    int r = idx / ldp, c = idx - r * ldp;
    float v = (c < cols) ? W[r * cols + c] : 0.0f;
    Wp[(((unsigned)r >> 1) * ldp + c) * 2 + (r & 1)] = v;
}

// x_proj = x @ W_proj  [N,128]x[128,4]
__global__ void xproj_kernel(const float* __restrict__ x,
                             const float* __restrict__ Wp,
                             float* __restrict__ xp, int n) {
    int i = blockIdx.x * blockDim.x + threadIdx.x;
    if (i >= n) return;
    const float* xr = x + (size_t)i * NFEAT;
    float a0 = 0.f, a1 = 0.f, a2 = 0.f, a3 = 0.f;
#pragma unroll 8
    for (int k = 0; k < NFEAT; ++k) {
        float xv = xr[k];
        float4 w = *(const float4*)(Wp + k * 4);
        a0 += xv * w.x; a1 += xv * w.y; a2 += xv * w.z; a3 += xv * w.w;
    }
    *(float4*)(xp + (size_t)i * 4) = make_float4(a0, a1, a2, a3);
}

// -------------------------------------------------- WMMA f32 16x16x4 GEMM
// C[M x 144] = A[M x K] * B[K x 144]   (B zero-padded + K-pair interleaved,
// K % 4 == 0, M % 16 == 0). Block = 8 waves; wave w owns m-tile
// (blockIdx.x*8 + w) and all 9 n-tiles. B staged into LDS by the TDM.
__global__ __launch_bounds__(256) void gemm_wmma_f32_kernel(
        const float* __restrict__ A, int lda,
        const float* __restrict__ B,            // [K/2 x LD x 2] interleaved
        float* __restrict__ C, int ldc,
        int mtiles, int K) {
    __shared__ float Bs[H2 * LD];               // 132*144 floats = 76,032 B

#if defined(__gfx1250__) && __has_builtin(__builtin_amdgcn_tensor_load_to_lds)
    // --- TDM: one 2D descriptor over the contiguous K*144 floats (copy order
    // is linear, so the interleaved layout is preserved verbatim in LDS).
    if ((threadIdx.x >> 5) == 0) {
        unsigned long long ga = (unsigned long long)(uintptr_t)B;
        unsigned int lds_off = (unsigned int)(uintptr_t)&Bs[0];  // addr[31:0] = LDS offset
        u32x4 g0 = { 1u,                                    // count=1, no gather
                     lds_off,                               // lds_addr
                     (unsigned int)ga,                      // global_addr[31:0]
                     ((unsigned int)(ga >> 32) & 0x01FFFFFFu) | 0x80000000u }; // addr[56:32] | type=2
        i32x8 g1 = { (int)0x00020000u,                      // data_size = 4B, no mask/flags
                     (int)(144u << 16),                     // tensor_dim0 = 144 (bits 79:48)
                     (int)(((unsigned)K & 0xFFFFu) << 16),  // tensor_dim1 = K  (bits 111:80)
                     (int)(144u << 16),                     // tile_dim0 = 144  (bits 127:112)
                     K,                                     // tile_dim1 = K; tile_dim2 = 0
                     144,                                   // tensor_dim0_stride = 144
                     0, 0 };
        i32x4 gz4 = { 0, 0, 0, 0 };                         // groups 2/3 unused (2D tensor)
        i32x8 gz8 = { 0, 0, 0, 0, 0, 0, 0, 0 };
        __builtin_amdgcn_tensor_load_to_lds(g0, g1, gz4, gz4, gz8, 0);
        __builtin_amdgcn_s_wait_tensorcnt(0);
    }
#else
    for (int i = threadIdx.x * 4; i < K * LD; i += blockDim.x * 4)
        *(float4*)&Bs[i] = *(const float4*)&B[i];
#endif
    __syncthreads();

    int wid  = threadIdx.x >> 5;
    int lane = threadIdx.x & 31;
    int tm   = blockIdx.x * 8 + wid;
    if (tm >= mtiles) return;                   // wave-uniform: EXEC stays all-1s

    int hi  = lane >> 4;                        // 0: lanes 0-15, 1: lanes 16-31
    int l15 = lane & 15;
    int kg  = hi << 1;                          // K sub-offset 0 or 2

    const float* arow = A + (size_t)(tm * 16 + l15) * lda + kg;

    v8f acc[NT] = {};
    for (int k = 0; k < K; k += 4) {
        v2f a = *(const v2f*)(arow + k);        // rows m, K = {k+kg, k+kg+1}
        // interleaved B: pair {k+kg, k+kg+1} of column c at Bs[(((k+kg)>>1)*LD + c)*2]
        const float* bbase = &Bs[((((unsigned)(k + kg)) >> 1) * LD + l15) * 2];
#pragma unroll
        for (int t = 0; t < NT; ++t) {
            v2f b = *(const v2f*)(bbase + t * 32);   // single ds_load_b64
            acc[t] = __builtin_amdgcn_wmma_f32_16x16x4_f32(
                false, a, false, b, (short)0, acc[t], false, false);
        }
    }

    // C/D layout: lanes 0-15 -> M = r, lanes 16-31 -> M = r + 8 ; N = l15
    float* cbase = C + (size_t)(tm * 16 + (hi << 3)) * ldc + l15;
#pragma unroll
    for (int t = 0; t < NT; ++t)
#pragma unroll
        for (int r = 0; r < 8; ++r)
            cbase[(size_t)r * ldc + t * 16] = acc[t][r];
}

// ----------------------------------------------------------- edge scatter
// grid.x covers edges, grid.y covers 4-feature chunks (L2-resident atomics)
__global__ void scatter_kernel(const float* __restrict__ h,
                               float* __restrict__ agg,
                               const int* __restrict__ row,
                               const int* __restrict__ col,
                               const float* __restrict__ coef, int E) {
    int e = blockIdx.x * blockDim.x + threadIdx.x;
    if (e >= E) return;
    int f = blockIdx.y << 2;
    int r = row[e], c = col[e];
    float cf = coef[e];
    float4 v = *(const float4*)(h + (size_t)r * LD + f);
    float* dst = agg + (size_t)c * LD + f;
    atomic_add_f32(dst + 0, v.x * cf);
    atomic_add_f32(dst + 1, v.y * cf);
    atomic_add_f32(dst + 2, v.z * cf);
    atomic_add_f32(dst + 3, v.w * cf);
}

// ------------------------------------------------ conv1 epilogue -> hcat
__global__ void finish1_kernel(const float* __restrict__ agg,
                               const float* __restrict__ h,
                               const float* __restrict__ invdeg,
                               const float* __restrict__ b1,
                               const float* __restrict__ xp,
                               float* __restrict__ hcat, int n) {
    int idx = blockIdx.x * blockDim.x + threadIdx.x;   // n * 36 chunks
    if (idx >= n * 36) return;
    int i = idx / 36;
    int f = (idx - i * 36) << 2;
    float4 o;
    if (f < NFEAT) {
        float id  = invdeg[i];
        float4 ag = *(const float4*)(agg + (size_t)i * LD + f);
        float4 hv = *(const float4*)(h + (size_t)i * LD + f);
        float4 bb = *(const float4*)(b1 + f);
        o.x = fmaxf(ag.x + hv.x * id + bb.x, 0.f);
        o.y = fmaxf(ag.y + hv.y * id + bb.y, 0.f);
        o.z = fmaxf(ag.z + hv.z * id + bb.z, 0.f);
        o.w = fmaxf(ag.w + hv.w * id + bb.w, 0.f);
    } else if (f == NFEAT) {
        o = *(const float4*)(xp + (size_t)i * 4);
    } else {
        o = make_float4(0.f, 0.f, 0.f, 0.f);
    }
    *(float4*)(hcat + (size_t)i * LD + f) = o;
}

// ------------------------------------------------ conv2 epilogue -> out
__global__ void finish2_kernel(const float* __restrict__ agg,
                               const float* __restrict__ g,
                               const float* __restrict__ invdeg,
                               const float* __restrict__ b2,
                               const float* __restrict__ xp,
                               float* __restrict__ out, int n) {
    int idx = blockIdx.x * blockDim.x + threadIdx.x;   // n * 34 chunks
    if (idx >= n * 34) return;
    int i = idx / 34;
    int f = (idx - i * 34) << 2;
    float4 o;
    if (f < H2) {
        float id  = invdeg[i];
        float4 ag = *(const float4*)(agg + (size_t)i * LD + f);
        float4 gv = *(const float4*)(g + (size_t)i * LD + f);
        float4 bb = *(const float4*)(b2 + f);
        o.x = ag.x + gv.x * id + bb.x;
        o.y = ag.y + gv.y * id + bb.y;
        o.z = ag.z + gv.z * id + bb.z;
        o.w = ag.w + gv.w * id + bb.w;
    } else {
        o = *(const float4*)(xp + (size_t)i * 4);
    }
    *(float4*)(out + (size_t)i * OUTC + f) = o;
}

// ---------------------------------------------------------------- launcher
extern "C" void kernel_launch(void* const* d_in, const int* in_sizes, int n_in,
                              void* d_out, int out_size, void* d_ws, size_t ws_size,
                              hipStream_t stream) {
    const int*   edge  = (const int*)d_in[0];    // [2, E]
    const float* x     = (const float*)d_in[1];  // [N, 128]
    const float* Wproj = (const float*)d_in[2];  // [128, 4]
    const float* W1    = (const float*)d_in[3];  // [128, 128]
    const float* b1    = (const float*)d_in[4];  // [128]
    const float* W2    = (const float*)d_in[5];  // [132, 132]
    const float* b2    = (const float*)d_in[6];  // [132]
    float*       out   = (float*)d_out;

    const int E = in_sizes[0] / 2;
    const int N = in_sizes[1] / NFEAT;
    const int* row = edge;
    const int* col = edge + E;

    // workspace carve-up (floats)
    float* ws   = (float*)d_ws;
    size_t big  = (size_t)N * LD;                // 7.2M floats each
    float* bufA = ws;                            // h1, then g2
    float* bufB = bufA + big;                    // agg1, then agg2
    float* bufC = bufB + big;                    // hcat
    float* dinv = bufC + big;                    // N (deg, then rsqrt in place)
    float* invd = dinv + N;                      // N
    float* xp   = invd + N;                      // N*4
    float* Wp1  = xp + (size_t)N * 4;            // 128*144 (interleaved)
    float* Wp2  = Wp1 + NFEAT * LD;              // 132*144 (interleaved)
    float* coef = Wp2 + H2 * LD;                 // E

    const int B256 = 256;
    auto blocks = [](long long n, int b) { return (unsigned)((n + b - 1) / b); };

    // 1) degrees + per-edge coefficients
    zero_f4_kernel<<<blocks(N / 4, B256), B256, 0, stream>>>((float4*)dinv, N / 4);
    degree_count_kernel<<<blocks(E, B256), B256, 0, stream>>>(col, dinv, E);
    degree_fin_kernel<<<blocks(N, B256), B256, 0, stream>>>(dinv, invd, N);
    coef_kernel<<<blocks(E, B256), B256, 0, stream>>>(row, col, dinv, coef, E);

    // 2) x_proj
    xproj_kernel<<<blocks(N, B256), B256, 0, stream>>>(x, Wproj, xp, N);

    // 3) pad + interleave weights
    pad_w_kernel<<<blocks(NFEAT * LD, B256), B256, 0, stream>>>(W1, Wp1, NFEAT, NFEAT, LD);
    pad_w_kernel<<<blocks(H2 * LD, B256), B256, 0, stream>>>(W2, Wp2, H2, H2, LD);

    // 4) h1 = x @ W1p   (3125 m-tiles, 8 per block)
    {
        int mt = N / 16;
        gemm_wmma_f32_kernel<<<blocks((long long)mt, 8), B256, 0, stream>>>(
            x, NFEAT, Wp1, bufA, LD, mt, NFEAT);
    }

    // 5) conv1 aggregate
    zero_f4_kernel<<<blocks(big / 4, B256), B256, 0, stream>>>((float4*)bufB, (int)(big / 4));
    {
        dim3 g(blocks(E, B256), NFEAT / 4);     // 32 feature chunks
        scatter_kernel<<<g, B256, 0, stream>>>(bufA, bufB, row, col, coef, E);
    }
    finish1_kernel<<<blocks((long long)N * 36, B256), B256, 0, stream>>>(
        bufB, bufA, invd, b1, xp, bufC, N);

    // 6) g2 = hcat @ W2p   (K=132)
    {
        int mt = N / 16;
        gemm_wmma_f32_kernel<<<blocks((long long)mt, 8), B256, 0, stream>>>(
            bufC, LD, Wp2, bufA, LD, mt, H2);
    }

    // 7) conv2 aggregate
    zero_f4_kernel<<<blocks(big / 4, B256), B256, 0, stream>>>((float4*)bufB, (int)(big / 4));
    {
        dim3 g(blocks(E, B256), H2 / 4);        // 33 feature chunks
        scatter_kernel<<<g, B256, 0, stream>>>(bufA, bufB, row, col, coef, E);
    }
    finish2_kernel<<<blocks((long long)N * 34, B256), B256, 0, stream>>>(
        bufB, bufA, invd, b2, xp, out, N);
}